// TPSAlign_4922032521570
// MI455X (gfx1250) — compile-verified
//
#include <hip/hip_runtime.h>
#include <hip/hip_bf16.h>
#include <stdint.h>

// Problem constants (fixed by the reference)
#define FM_B 8
#define FM_C 128
#define FM_H 64
#define FM_W 160
#define G_H  32
#define G_W  100

#define HW        (FM_H * FM_W)      // 10240 floats per (b,c) plane
#define PLANE_B   (HW * 4)           // 40960 bytes per plane
#define PTS       (G_H * G_W)        // 3200 sample points per n
#define CB        4                  // channel blocks per n
#define C_PER     (FM_C / CB)        // 32 channels per workgroup
#define BLK       1024               // threads per workgroup (32 wave32s)
#define SLOTS     4                  // ceil(PTS / BLK) point slots per thread

// Issue one plane (40960 B) as coalesced async global->LDS B128 copies.
// IOFFSET is added to BOTH the LDS and global addresses, so one base pair
// covers all three rounds. Rounds 0/1 full (1024 lanes x 16B = 16KB each);
// round 2 covers the remaining 8KB (512 lanes).
__device__ __forceinline__ void async_load_plane(float* dst, const float* src,
                                                 int t)
{
    unsigned    lds = (unsigned)(uintptr_t)dst + (unsigned)(t * 16);
    const char* g   = (const char*)src + (size_t)(t * 16);
    asm volatile("global_load_async_to_lds_b128 %0, %1, off"
                 :: "v"(lds), "v"(g) : "memory");
    asm volatile("global_load_async_to_lds_b128 %0, %1, off offset:16384"
                 :: "v"(lds), "v"(g) : "memory");
    if (t < (PLANE_B - 32768) / 16) {   // t < 512
        asm volatile("global_load_async_to_lds_b128 %0, %1, off offset:32768"
                     :: "v"(lds), "v"(g) : "memory");
    }
}

// One workgroup = one n (all 3200 points) x a 32-channel slice.
// Double-buffered plane staging: while gathering channel c from LDS, the
// async engine streams channel c+1 into the other buffer.
__global__ __launch_bounds__(BLK) void grid_sample_staged(
    const float* __restrict__ fm,     // [B,C,H,W]
    const float* __restrict__ grids,  // [N,GH,GW,2]
    const int*   __restrict__ bidx,   // [N]
    float*       __restrict__ out)    // [N,C,GH,GW]
{
    extern __shared__ float lbuf[];   // 2 x HW floats = 80 KB

    const int t    = threadIdx.x;
    const int n    = blockIdx.x / CB;
    const int cblk = blockIdx.x - n * CB;
    const int b    = bidx[n];

    const float* __restrict__ fmb =
        fm + ((size_t)b * FM_C + (size_t)cblk * C_PER) * HW;

    // ---- per-point state, computed once:
    //   offT/offB: row-pair base offsets (top/bottom rows)
    //   wTl,wTh,wBl,wBh: weights for elements [off], [off+1] of each row
    float wTl[SLOTS], wTh[SLOTS], wBl[SLOTS], wBh[SLOTS];
    int   offT[SLOTS], offB[SLOTS];
    int   pidx[SLOTS];

#pragma unroll
    for (int s = 0; s < SLOTS; ++s) {
        const int p = s * BLK + t;
        pidx[s] = (p < PTS) ? p : -1;
        wTl[s] = wTh[s] = wBl[s] = wBh[s] = 0.0f;
        offT[s] = offB[s] = 0;
        if (p < PTS) {
            const float gx = grids[((size_t)n * PTS + p) * 2 + 0];
            const float gy = grids[((size_t)n * PTS + p) * 2 + 1];

            const float x = (gx + 1.0f) * (0.5f * FM_W) - 0.5f;
            const float y = (gy + 1.0f) * (0.5f * FM_H) - 0.5f;

            const float x0f = floorf(x);
            const float y0f = floorf(y);
            const int x0 = (int)x0f, y0 = (int)y0f;
            const int x1 = x0 + 1,  y1 = y0 + 1;
            const float fx = x - x0f, fy = y - y0f;

            // separable weights with validity folded in
            // (== reference's  value*valid*w  for finite data)
            const float ax0 = (1.0f - fx) *
                (float)((x0 >= 0) & (x0 <= FM_W - 1));
            const float ax1 = fx *
                (float)((x1 >= 0) & (x1 <= FM_W - 1));
            const float ay0 = (1.0f - fy) *
                (float)((y0 >= 0) & (y0 <= FM_H - 1));
            const float ay1 = fy *
                (float)((y1 >= 0) & (y1 <= FM_H - 1));

            // row base xb = clamp(x0, 0, W-2); remap x-weights onto the
            // adjacent element pair [xb], [xb+1] (edge cases exact because
            // the out-of-range weight was already zeroed)
            float wl, wh;
            if (x0 < 0)             { wl = ax1; wh = 0.0f; }  // left edge
            else if (x0 > FM_W - 2) { wl = 0.0f; wh = ax0; }  // right edge
            else                    { wl = ax0; wh = ax1; }
            const int xb  = min(max(x0, 0), FM_W - 2);
            const int yi0 = min(max(y0, 0), FM_H - 1);
            const int yi1 = min(max(y1, 0), FM_H - 1);

            offT[s] = yi0 * FM_W + xb;
            offB[s] = yi1 * FM_W + xb;
            wTl[s] = wl * ay0;  wTh[s] = wh * ay0;
            wBl[s] = wl * ay1;  wBh[s] = wh * ay1;
        }
    }

    // ---- double-buffered channel loop ----
    async_load_plane(&lbuf[0], fmb, t);          // preload channel 0

    for (int cl = 0; cl < C_PER; ++cl) {
        asm volatile("s_wait_asynccnt 0" ::: "memory");
        __syncthreads();   // buf[cl&1] fully written; prior reads of the
                           // other buffer complete on all waves

        if (cl + 1 < C_PER)
            async_load_plane(&lbuf[((cl + 1) & 1) * HW],
                             fmb + (size_t)(cl + 1) * HW, t);

        const float* __restrict__ L = &lbuf[(cl & 1) * HW];
        float* __restrict__ oc =
            out + ((size_t)n * FM_C + (size_t)cblk * C_PER + cl) * PTS;

#pragma unroll
        for (int s = 0; s < SLOTS; ++s) {
            if (pidx[s] >= 0) {
                // adjacent pairs -> single DS_LOAD_2ADDR_B32 per row
                const float eT0 = L[offT[s]];
                const float eT1 = L[offT[s] + 1];
                const float eB0 = L[offB[s]];
                const float eB1 = L[offB[s] + 1];
                const float v = eT0 * wTl[s] + eT1 * wTh[s] +
                                eB0 * wBl[s] + eB1 * wBh[s];
                // output is a 210MB one-shot stream: NT store keeps the
                // feature map resident in L2
                __builtin_nontemporal_store(v, oc + pidx[s]);
            }
        }
    }
}

__global__ void copy_texts_kernel(const long long* __restrict__ t,
                                  long long* __restrict__ o, int n)
{
    int i = blockIdx.x * blockDim.x + threadIdx.x;
    if (i < n) o[i] = t[i];
}

extern "C" void kernel_launch(void* const* d_in, const int* in_sizes, int n_in,
                              void* d_out, int out_size, void* d_ws, size_t ws_size,
                              hipStream_t stream)
{
    const float*     fm    = (const float*)d_in[0];     // [8,128,64,160] f32
    const float*     grids = (const float*)d_in[1];     // [N,32,100,2] f32
    const int*       bidx  = (const int*)d_in[2];       // [N] i32
    const long long* texts = (const long long*)d_in[3]; // [N,25] i64

    float* out = (float*)d_out;

    const int N = in_sizes[2];                  // 128

    const size_t shmem = 2 * (size_t)HW * sizeof(float);   // 80 KB
    dim3 grid(N * CB);                          // 512 workgroups
    dim3 block(BLK);                            // 1024 threads (32 waves)
    grid_sample_staged<<<grid, block, shmem, stream>>>(fm, grids, bidx, out);

    // texts pass-through appended after feats (N*C*GH*GW f32, 8-byte aligned)
    long long* otext = (long long*)(out + (size_t)N * FM_C * PTS);
    const int tcount = in_sizes[3];             // 3200 int64 elements
    copy_texts_kernel<<<(tcount + 255) / 256, 256, 0, stream>>>(texts, otext, tcount);
}